// FTDNNPronscorer_30717606101127
// MI455X (gfx1250) — compile-verified
//
#include <hip/hip_runtime.h>
#include <math.h>

#define BB 32
#define TT 4096
#define MM 256
#define DD 257   // feats dim (= x last dim; feats[d<256]=x[...,d+1], feats[256]=n2f)
#define OO 40
#define OP 48    // O padded to 3 WMMA n-tiles
#define DP 260   // feats dim padded to multiple of 4
#define NS 8     // K-split segments for the segment-sum GEMM
#define KSEG (TT / NS)  // 512

typedef float v2f __attribute__((ext_vector_type(2)));
typedef float v8f __attribute__((ext_vector_type(8)));

__device__ __forceinline__ v8f wmma_f32x4(v2f a, v2f b, v8f c) {
#if __has_builtin(__builtin_amdgcn_wmma_f32_16x16x4_f32)
  // 8 args: (neg_a, A, neg_b, B, c_mod, C, reuse_a, reuse_b)
  return __builtin_amdgcn_wmma_f32_16x16x4_f32(false, a, false, b, (short)0, c,
                                               false, false);
#else
  (void)a; (void)b;
  return c;
#endif
}

// ---------------------------------------------------------------------------
// Kernel 0: pad W (O x D) -> Wp (OP x DP), bias -> bp (OP)
// ---------------------------------------------------------------------------
__global__ void k_pad(const float* __restrict__ W, const float* __restrict__ bias,
                      float* __restrict__ Wp, float* __restrict__ bp) {
  int i = blockIdx.x * blockDim.x + threadIdx.x;
  if (i < OP * DP) {
    int o = i / DP, d = i % DP;
    Wp[i] = (o < OO && d < DD) ? W[o * DD + d] : 0.0f;
  }
  if (i < OP) bp[i] = (i < OO) ? bias[i] : 0.0f;
}

// ---------------------------------------------------------------------------
// Kernel 1: nfb[b,m] = sum_t cum[b,m,t]  (one wave per (b,m) row)
// ---------------------------------------------------------------------------
__global__ void k_nfb(const float* __restrict__ cum, float* __restrict__ nfb) {
  int wid  = (blockIdx.x * blockDim.x + threadIdx.x) >> 5;  // 0..B*M-1
  int lane = threadIdx.x & 31;
  const float* row = cum + (size_t)wid * TT;
  float s = 0.0f;
  for (int t = lane; t < TT; t += 32) s += row[t];
  for (int off = 16; off > 0; off >>= 1) s += __shfl_down(s, off, 32);
  if (lane == 0) nfb[wid] = s;
}

// ---------------------------------------------------------------------------
// Kernel 2: n2f[b,t] = sum_m cum[b,m,t] * nfb[b,m]
// ---------------------------------------------------------------------------
__global__ void k_n2f(const float* __restrict__ cum, const float* __restrict__ nfb,
                      float* __restrict__ n2f) {
  int t = blockIdx.x * blockDim.x + threadIdx.x;
  int b = blockIdx.y;
  const float* cb = cum + (size_t)b * MM * TT + t;  // column t, strided by TT
  const float* nb = nfb + b * MM;                   // uniform per block
  float s = 0.0f;
  for (int m = 0; m < MM; ++m) s += cb[(size_t)m * TT] * nb[m];
  n2f[b * TT + t] = s;
}

// ---------------------------------------------------------------------------
// Kernel 3: feature GEMM (WMMA f32 16x16x4), fused bias + |mask| + padded store
//   maskedp[b,t,o(48)] = (sum_d feats[b,t,d]*Wp[o,d] + bp[o]) * |mask[b,t,o]|
// One wave per 16-frame tile; grid is an exact multiple (EXEC all ones).
// ---------------------------------------------------------------------------
__global__ __launch_bounds__(256) void k_gemm1(
    const float* __restrict__ x, const float* __restrict__ mask,
    const float* __restrict__ n2f, const float* __restrict__ Wp,
    const float* __restrict__ bp, float* __restrict__ maskedp) {
  int lane = threadIdx.x & 31;
  int ft   = blockIdx.x * 8 + (threadIdx.x >> 5);  // frame-tile id, 0..B*T/16-1
  int b    = ft >> 8;                              // 256 tiles per batch
  int t0   = (ft & 255) << 4;
  int half = lane >> 4, l = lane & 15;

  const float* arow = x + ((size_t)(b * TT + t0 + l) * DD) + 1;
  const float* w0 = Wp + (size_t)(0 * 16 + l) * DP;
  const float* w1 = Wp + (size_t)(1 * 16 + l) * DP;
  const float* w2 = Wp + (size_t)(2 * 16 + l) * DP;

  v8f c0 = {}, c1 = {}, c2 = {};

  for (int k = 0; k < 256; k += 4) {
    int d = k + 2 * half;
    v2f a;  a.x  = arow[d];      a.y  = arow[d + 1];
    v2f b0; b0.x = w0[d];        b0.y = w0[d + 1];
    v2f b1; b1.x = w1[d];        b1.y = w1[d + 1];
    v2f b2; b2.x = w2[d];        b2.y = w2[d + 1];
    c0 = wmma_f32x4(a, b0, c0);
    c1 = wmma_f32x4(a, b1, c1);
    c2 = wmma_f32x4(a, b2, c2);
  }
  { // tail k=256: feats[256]=n2f, feats[257..259]=0 (Wp cols 257..259 are 0 too)
    int d = 256 + 2 * half;
    v2f a;  a.x  = (half == 0) ? n2f[b * TT + t0 + l] : 0.0f;  a.y = 0.0f;
    v2f b0; b0.x = w0[d]; b0.y = w0[d + 1];
    v2f b1; b1.x = w1[d]; b1.y = w1[d + 1];
    v2f b2; b2.x = w2[d]; b2.y = w2[d + 1];
    c0 = wmma_f32x4(a, b0, c0);
    c1 = wmma_f32x4(a, b1, c1);
    c2 = wmma_f32x4(a, b2, c2);
  }

  // Epilogue: C vgpr i holds row (i + 8*half), col l of the 16x16 tile.
  v8f acc[3] = {c0, c1, c2};
#pragma unroll
  for (int nt = 0; nt < 3; ++nt) {
    int o = nt * 16 + l;
    float bias = bp[o];
#pragma unroll
    for (int i = 0; i < 8; ++i) {
      int t = t0 + i + 8 * half;
      float mv = (o < OO) ? fabsf(mask[(size_t)(b * TT + t) * OO + o]) : 0.0f;
      maskedp[(size_t)(b * TT + t) * OP + o] = (acc[nt][i] + bias) * mv;
    }
  }
}

// ---------------------------------------------------------------------------
// Segment-sum GEMM inner body (shared by split and fallback kernels).
// Computes, for tile (b, m0) over k in [k0, k0+klen):
//   sacc[nt] += cum[b,m,:] (*) maskedp[b,:,:]   (summed partial)
//   facc[nt] += cum[b,m,:] (*) mask[b,:,:]      (counts partial)
// ---------------------------------------------------------------------------
__device__ __forceinline__ void seg_gemm_body(
    const float* __restrict__ cum, const float* __restrict__ mask,
    const float* __restrict__ maskedp, int b, int m0, int k0, int klen,
    int half, int l, v8f sacc[3], v8f facc[3]) {
  const float* arow = cum + (size_t)(b * MM + m0 + l) * TT;
  bool g2ok = (32 + l) < OO;

  for (int k = k0; k < k0 + klen; k += 4) {
    int kk = k + 2 * half;
    v2f a; a.x = arow[kk]; a.y = arow[kk + 1];

    const float* mp = maskedp + (size_t)(b * TT + kk) * OP;  // padded, stride 48
    v2f b0; b0.x = mp[l];          b0.y = mp[OP + l];
    v2f b1; b1.x = mp[16 + l];     b1.y = mp[OP + 16 + l];
    v2f b2; b2.x = mp[32 + l];     b2.y = mp[OP + 32 + l];

    const float* mk = mask + (size_t)(b * TT + kk) * OO;     // unpadded, stride 40
    v2f g0; g0.x = mk[l];          g0.y = mk[OO + l];
    v2f g1; g1.x = mk[16 + l];     g1.y = mk[OO + 16 + l];
    v2f g2; g2.x = g2ok ? mk[32 + l] : 0.0f;
            g2.y = g2ok ? mk[OO + 32 + l] : 0.0f;

    sacc[0] = wmma_f32x4(a, b0, sacc[0]);
    sacc[1] = wmma_f32x4(a, b1, sacc[1]);
    sacc[2] = wmma_f32x4(a, b2, sacc[2]);
    facc[0] = wmma_f32x4(a, g0, facc[0]);
    facc[1] = wmma_f32x4(a, g1, facc[1]);
    facc[2] = wmma_f32x4(a, g2, facc[2]);
  }
}

// ---------------------------------------------------------------------------
// Kernel 4a (preferred): K-split partial segment-sum GEMM.
// 4096 waves (8 segments x 512 tiles) for memory-latency coverage.
// Wave id = seg*512 + tile, so the 8 waves of a block share one segment and
// consecutive m-tiles -> identical B-fragment cachelines (L1/L2 reuse).
// ---------------------------------------------------------------------------
__global__ __launch_bounds__(256) void k_gemm2_part(
    const float* __restrict__ cum, const float* __restrict__ mask,
    const float* __restrict__ maskedp, float* __restrict__ part_s,
    float* __restrict__ part_f) {
  int lane = threadIdx.x & 31;
  int wid  = blockIdx.x * 8 + (threadIdx.x >> 5);  // 0..NS*B*16-1
  int seg  = wid >> 9;           // 512 tiles per segment
  int tile = wid & 511;
  int b    = tile >> 4;
  int m0   = (tile & 15) << 4;
  int half = lane >> 4, l = lane & 15;

  v8f sacc[3] = {};
  v8f facc[3] = {};
  seg_gemm_body(cum, mask, maskedp, b, m0, seg * KSEG, KSEG, half, l, sacc, facc);

  size_t segbase = (size_t)seg * BB * MM * OP;
#pragma unroll
  for (int nt = 0; nt < 3; ++nt) {
    int o = nt * 16 + l;
#pragma unroll
    for (int i = 0; i < 8; ++i) {
      int m = m0 + i + 8 * half;
      size_t idx = segbase + (size_t)(b * MM + m) * OP + o;
      part_s[idx] = sacc[nt][i];
      part_f[idx] = facc[nt][i];
    }
  }
}

// Finalize: sum NS partials, counts==0 -> 1, divide.
__global__ void k_fin(const float* __restrict__ part_s,
                      const float* __restrict__ part_f, float* __restrict__ out) {
  int i = blockIdx.x * blockDim.x + threadIdx.x;  // 0..B*M*OO-1
  if (i >= BB * MM * OO) return;
  int o  = i % OO;
  int bm = i / OO;
  size_t base = (size_t)bm * OP + o;
  float s = 0.0f, f = 0.0f;
#pragma unroll
  for (int seg = 0; seg < NS; ++seg) {
    s += part_s[(size_t)seg * BB * MM * OP + base];
    f += part_f[(size_t)seg * BB * MM * OP + base];
  }
  f = (f == 0.0f) ? 1.0f : f;
  out[i] = s / f;
}

// ---------------------------------------------------------------------------
// Kernel 4b (fallback if workspace too small): single-pass, 512 waves.
// ---------------------------------------------------------------------------
__global__ __launch_bounds__(256) void k_gemm2_full(
    const float* __restrict__ cum, const float* __restrict__ mask,
    const float* __restrict__ maskedp, float* __restrict__ out) {
  int lane = threadIdx.x & 31;
  int wid  = blockIdx.x * 8 + (threadIdx.x >> 5);  // 0..B*16-1
  int b    = wid >> 4;
  int m0   = (wid & 15) << 4;
  int half = lane >> 4, l = lane & 15;

  v8f sacc[3] = {};
  v8f facc[3] = {};
  seg_gemm_body(cum, mask, maskedp, b, m0, 0, TT, half, l, sacc, facc);

#pragma unroll
  for (int nt = 0; nt < 3; ++nt) {
    int o = nt * 16 + l;
    if (o < OO) {
#pragma unroll
      for (int i = 0; i < 8; ++i) {
        int m = m0 + i + 8 * half;
        float fc = facc[nt][i];
        fc = (fc == 0.0f) ? 1.0f : fc;
        out[(size_t)(b * MM + m) * OO + o] = sacc[nt][i] / fc;
      }
    }
  }
}

// ---------------------------------------------------------------------------
extern "C" void kernel_launch(void* const* d_in, const int* in_sizes, int n_in,
                              void* d_out, int out_size, void* d_ws, size_t ws_size,
                              hipStream_t stream) {
  const float* x    = (const float*)d_in[0];  // (B,T,257)
  const float* mask = (const float*)d_in[1];  // (B,T,40)
  const float* cum  = (const float*)d_in[2];  // (B,M,T)
  const float* W    = (const float*)d_in[3];  // (40,257)
  const float* bias = (const float*)d_in[4];  // (40,)
  float* out = (float*)d_out;                 // (B,M,40)

  // Workspace layout (floats):
  //   Wp | bp | nfb | n2f | maskedp | [part_s | part_f]
  float* ws      = (float*)d_ws;
  float* Wp      = ws;                         // OP*DP         = 12480
  float* bp      = Wp + OP * DP;               // OP            = 48
  float* nfb     = bp + OP;                    // B*M           = 8192
  float* n2f     = nfb + BB * MM;              // B*T           = 131072
  float* maskedp = n2f + BB * TT;              // B*T*OP        = 6291456
  float* part_s  = maskedp + (size_t)BB * TT * OP;  // NS*B*M*OP = 3145728
  float* part_f  = part_s + (size_t)NS * BB * MM * OP;

  size_t base_floats  = (size_t)(OP * DP + OP + BB * MM) + (size_t)BB * TT +
                        (size_t)BB * TT * OP;
  size_t split_floats = base_floats + 2ull * NS * BB * MM * OP;

  k_pad<<<(OP * DP + 255) / 256, 256, 0, stream>>>(W, bias, Wp, bp);
  k_nfb<<<(BB * MM * 32) / 256, 256, 0, stream>>>(cum, nfb);
  k_n2f<<<dim3(TT / 256, BB), 256, 0, stream>>>(cum, nfb, n2f);
  k_gemm1<<<(BB * TT / 16) / 8, 256, 0, stream>>>(x, mask, n2f, Wp, bp, maskedp);

  if (ws_size >= split_floats * sizeof(float)) {
    // 4096 waves: enough memory-level parallelism to stream cum (134 MB)
    // from HBM at full bandwidth.
    k_gemm2_part<<<(NS * BB * MM / 16) / 8, 256, 0, stream>>>(cum, mask, maskedp,
                                                              part_s, part_f);
    k_fin<<<(BB * MM * OO + 255) / 256, 256, 0, stream>>>(part_s, part_f, out);
  } else {
    k_gemm2_full<<<(BB * MM / 16) / 8, 256, 0, stream>>>(cum, mask, maskedp, out);
  }
}